// SplineLinear_5342939316890
// MI455X (gfx1250) — compile-verified
//
#include <hip/hip_runtime.h>

typedef __attribute__((ext_vector_type(2))) float v2f;
typedef __attribute__((ext_vector_type(4))) float v4f;
typedef __attribute__((ext_vector_type(8))) float v8f;

#define BATCH 16384
#define IN_F  4096
#define OUT_F 4096

// Keys cubic convolution weights, a = -0.75
__device__ __forceinline__ float cub_inner(float s) {   // |s| <= 1
    return ((1.25f * s - 2.25f) * s) * s + 1.0f;        // ((a+2)s-(a+3))s^2+1
}
__device__ __forceinline__ float cub_outer(float s) {   // 1 < |s| < 2
    return ((-0.75f * s + 3.75f) * s - 6.0f) * s + 3.0f; // ((as-5a)s+8a)s-4a
}

// Kernel 0: build G = A_in @ CP^T (SoA: G[h*4096+i]), A_out rows (AoS), bias.
// Note IN_F == OUT_F and CP_H == CP_W == CP_B == 4, so one cubic row serves both.
__global__ void spline_setup(const float* __restrict__ wcp,
                             const float* __restrict__ bcp,
                             float* __restrict__ G,
                             float* __restrict__ Aout,
                             float* __restrict__ bias) {
    int i = blockIdx.x * blockDim.x + threadIdx.x;
    if (i >= OUT_F) return;
    float x  = (float)i * (3.0f / 4095.0f);   // align_corners=True source coord
    float x0 = floorf(x);
    float t  = x - x0;
    int x0i  = (int)x0;

    float wv[4] = { cub_outer(t + 1.0f), cub_inner(t),
                    cub_inner(1.0f - t), cub_outer(2.0f - t) };
    float ain[4] = {0.f, 0.f, 0.f, 0.f};
#pragma unroll
    for (int k = 0; k < 4; ++k) {             // taps at x0-1..x0+2, clamped
        int idx = x0i + k - 1;
        idx = idx < 0 ? 0 : (idx > 3 ? 3 : idx);
        ain[idx] += wv[k];
    }
#pragma unroll
    for (int h = 0; h < 4; ++h) {
        Aout[i * 4 + h] = ain[h];
        float g = 0.f;
#pragma unroll
        for (int w = 0; w < 4; ++w) g += wcp[h * 4 + w] * ain[w];
        G[h * IN_F + i] = g;
    }
    // linear bias interp, align_corners=True, same coord
    int lo = x0i;
    int hi = (lo + 1 > 3) ? 3 : lo + 1;
    bias[i] = (1.0f - t) * bcp[lo] + t * bcp[hi];
}

// Kernel 1: coef[b][h] = sum_i x[b][i] * G[h][i].  One wave per row, G in LDS.
// x is streamed once (256 MB) -> non-temporal loads keep it out of L2.
__global__ void __launch_bounds__(256)
spline_reduce(const float* __restrict__ x,
              const float* __restrict__ G,
              float* __restrict__ coef) {
    __shared__ float sG[4 * IN_F];            // 64 KB
    for (int idx = threadIdx.x; idx < 4 * IN_F; idx += 256)
        sG[idx] = G[idx];
    __syncthreads();

    int wave = threadIdx.x >> 5;
    int lane = threadIdx.x & 31;
    int b    = blockIdx.x * 8 + wave;
    const float* xr = x + (size_t)b * IN_F;

    float a0 = 0.f, a1 = 0.f, a2 = 0.f, a3 = 0.f;
    for (int i = lane * 4; i < IN_F; i += 128) {
        v4f xv = __builtin_nontemporal_load((const v4f*)(xr + i));
        v4f g0 = *(const v4f*)(sG + 0 * IN_F + i);
        v4f g1 = *(const v4f*)(sG + 1 * IN_F + i);
        v4f g2 = *(const v4f*)(sG + 2 * IN_F + i);
        v4f g3 = *(const v4f*)(sG + 3 * IN_F + i);
        a0 += xv.x * g0.x + xv.y * g0.y + xv.z * g0.z + xv.w * g0.w;
        a1 += xv.x * g1.x + xv.y * g1.y + xv.z * g1.z + xv.w * g1.w;
        a2 += xv.x * g2.x + xv.y * g2.y + xv.z * g2.z + xv.w * g2.w;
        a3 += xv.x * g3.x + xv.y * g3.y + xv.z * g3.z + xv.w * g3.w;
    }
#pragma unroll
    for (int off = 16; off > 0; off >>= 1) {  // wave32 butterfly reduce
        a0 += __shfl_xor(a0, off, 32);
        a1 += __shfl_xor(a1, off, 32);
        a2 += __shfl_xor(a2, off, 32);
        a3 += __shfl_xor(a3, off, 32);
    }
    if (lane == 0) {
        v4f c = {a0, a1, a2, a3};
        *(v4f*)(coef + (size_t)b * 4) = c;
    }
}

// Kernel 2: y[16x16 tile] = coef_tile(16x4) x Aout^T(4x16) + bias, via
// V_WMMA_F32_16X16X4_F32.  One wave handles 16 rows x 512 cols.
// A-operand layout (ISA 7.12.2, 32-bit 16x4): lanes 0-15 hold M, VGPR0=K0/K2,
// VGPR1=K1/K3 -> per-lane float2 load from row-major coef.  B (4x16) mirrors
// it on row-major Aout.  C/D: VGPR r = rows r / r+8, col = lane&15.
// y is written once (256 MB) -> non-temporal stores.
__global__ void __launch_bounds__(256)
spline_expand(const float* __restrict__ coef,
              const float* __restrict__ Aout,
              const float* __restrict__ bias,
              float* __restrict__ y) {
    int wid       = blockIdx.x * 8 + (threadIdx.x >> 5);
    int lane      = threadIdx.x & 31;
    int row_tile  = wid >> 3;        // 0..1023
    int col_chunk = wid & 7;         // 0..7  (512 cols each)
    int b0        = row_tile * 16;
    int lhalf     = lane >> 4;       // 0: K=0,1   1: K=2,3
    int llo       = lane & 15;

    v2f A = *(const v2f*)(coef + (size_t)(b0 + llo) * 4 + lhalf * 2);

    for (int t = 0; t < 32; ++t) {
        int o0 = col_chunk * 512 + t * 16;
        v2f B = *(const v2f*)(Aout + (size_t)(o0 + llo) * 4 + lhalf * 2);
        float bb = bias[o0 + llo];
        v8f c;
#pragma unroll
        for (int r = 0; r < 8; ++r) c[r] = bb;   // bias depends on N only

        v8f d = __builtin_amdgcn_wmma_f32_16x16x4_f32(
            /*neg_a=*/false, A, /*neg_b=*/false, B,
            /*c_mod=*/(short)0, c, /*reuse_a=*/false, /*reuse_b=*/false);

        float* yp = y + (size_t)(b0 + lhalf * 8) * OUT_F + o0 + llo;
#pragma unroll
        for (int r = 0; r < 8; ++r)
            __builtin_nontemporal_store(d[r], yp + (size_t)r * OUT_F);
    }
}

extern "C" void kernel_launch(void* const* d_in, const int* in_sizes, int n_in,
                              void* d_out, int out_size, void* d_ws, size_t ws_size,
                              hipStream_t stream) {
    const float* x   = (const float*)d_in[0];
    const float* wcp = (const float*)d_in[1];
    const float* bcp = (const float*)d_in[2];
    float* y  = (float*)d_out;

    float* ws   = (float*)d_ws;
    float* G    = ws;                    // 4*4096
    float* Aout = G + 4 * IN_F;          // 4096*4
    float* bias = Aout + 4 * OUT_F;      // 4096
    float* coef = bias + OUT_F;          // 16384*4  (total 400 KB)

    spline_setup <<<OUT_F / 256, 256, 0, stream>>>(wcp, bcp, G, Aout, bias);
    spline_reduce<<<BATCH / 8,   256, 0, stream>>>(x, G, coef);
    spline_expand<<<1024,        256, 0, stream>>>(coef, Aout, bias, y);
}